// GatedTransformerXLLayer_68788196212850
// MI455X (gfx1250) — compile-verified
//
#include <hip/hip_runtime.h>

// ---------------------------------------------------------------------------
// Gated Transformer-XL layer for MI455X (gfx1250), f32 end-to-end.
// All matmuls use V_WMMA_F32_16X16X4_F32 (exact f32, matches reference math).
// GEMM: 128x64 block tile, double-buffered LDS, k-pair-interleaved B layout
// so every WMMA B-fragment is a single aligned ds_load_b64 (no v_mov fixup).
// ---------------------------------------------------------------------------

#define CSZ 512
#define PSZ 512
#define BQ  8
#define DM  1024
#define NH  16
#define HDD 64
#define FSZ 1024
#define HID 4096

typedef float v2f __attribute__((ext_vector_type(2)));
typedef float v8f __attribute__((ext_vector_type(8)));

#define WMMA4(a, b, c) \
    __builtin_amdgcn_wmma_f32_16x16x4_f32(false, (a), false, (b), (short)0, (c), false, false)

#define GF_RELU 1
#define GF_ACC  2

// pair-row stride: 160 floats => +1 pair-row shifts banks by 32 (mod 64), so
// the two half-waves (adjacent k-pairs) occupy disjoint bank halves. 160*4 is
// 16B-aligned for float4/float2 accesses.
#define PSTRIDE 160

// ---------------------------------------------------------------------------
// Generic GEMM: C[M,N] = act( A[M,K] @ W[K,N] + bias  (+ C if GF_ACC) )
// Block: 256 threads = 8 waves. Tile 128x64, BK = 32, double buffered.
// Wave w owns rows w*16..w*16+16 and all 64 cols -> 4 WMMA accumulators;
// each A-fragment is reused across 4 WMMAs.
// ---------------------------------------------------------------------------
__launch_bounds__(256)
__global__ void gemm_wmma_kernel(const float* __restrict__ A,
                                 const float* __restrict__ W,
                                 const float* __restrict__ bias,
                                 float* __restrict__ C,
                                 int M, int N, int K, int flags)
{
    __shared__ float As[2][128][36];        // row-major A tile (stride 36: 16B aligned, conflict-free)
    __shared__ float WsP[2][16][PSTRIDE];   // W tile, k-pair interleaved: [k>>1][2n + (k&1)]

    const int tid  = threadIdx.x;
    const int lane = tid & 31;
    const int wave = tid >> 5;
    const int hf   = lane >> 4;     // half-wave: K pair {0,1} vs {2,3}
    const int l16  = lane & 15;
    const int m0   = blockIdx.y * 128;
    const int n0   = blockIdx.x * 64;

    const int rowA = tid >> 3;              // 0..31 (+32*it)
    const int c4A  = (tid & 7) << 2;
    const int rowW = tid >> 4;              // 0..15 (+16)
    const int c4W  = (tid & 15) << 2;

    float4 ra0, ra1, ra2, ra3, rw0, rw1;

#define LOAD_REGS(K0)  do {                                                   \
        const float* Ab = A + (size_t)m0 * K + (K0);                          \
        ra0 = *(const float4*)(Ab + (size_t)(rowA      ) * K + c4A);          \
        ra1 = *(const float4*)(Ab + (size_t)(rowA + 32 ) * K + c4A);          \
        ra2 = *(const float4*)(Ab + (size_t)(rowA + 64 ) * K + c4A);          \
        ra3 = *(const float4*)(Ab + (size_t)(rowA + 96 ) * K + c4A);          \
        const float* Wb = W + (size_t)(K0) * N + n0;                          \
        rw0 = *(const float4*)(Wb + (size_t)(rowW     ) * N + c4W);           \
        rw1 = *(const float4*)(Wb + (size_t)(rowW + 16) * N + c4W);           \
    } while (0)

#define STORE_TILE(BI)  do {                                                  \
        *(float4*)&As[BI][rowA      ][c4A] = ra0;                             \
        *(float4*)&As[BI][rowA + 32 ][c4A] = ra1;                             \
        *(float4*)&As[BI][rowA + 64 ][c4A] = ra2;                             \
        *(float4*)&As[BI][rowA + 96 ][c4A] = ra3;                             \
        {                                                                     \
            int pr = rowW >> 1, pb = rowW & 1;                                \
            WsP[BI][pr][2*(c4W+0) + pb] = rw0.x;                              \
            WsP[BI][pr][2*(c4W+1) + pb] = rw0.y;                              \
            WsP[BI][pr][2*(c4W+2) + pb] = rw0.z;                              \
            WsP[BI][pr][2*(c4W+3) + pb] = rw0.w;                              \
            int pr2 = (rowW + 16) >> 1;                                       \
            WsP[BI][pr2][2*(c4W+0) + pb] = rw1.x;                             \
            WsP[BI][pr2][2*(c4W+1) + pb] = rw1.y;                             \
            WsP[BI][pr2][2*(c4W+2) + pb] = rw1.z;                             \
            WsP[BI][pr2][2*(c4W+3) + pb] = rw1.w;                             \
        }                                                                     \
    } while (0)

    v8f acc[4];
    #pragma unroll
    for (int t = 0; t < 4; ++t) acc[t] = (v8f){0.f,0.f,0.f,0.f,0.f,0.f,0.f,0.f};

    // prologue: stage chunk 0
    LOAD_REGS(0);
    STORE_TILE(0);
    __syncthreads();

    const int nch = K >> 5;
    const int ar  = wave * 16 + l16;
    for (int c = 0; c < nch; ++c) {
        const int cur = c & 1;
        const bool more = (c + 1 < nch);
        if (more) {
            LOAD_REGS((c + 1) << 5);
            if (c + 2 < nch) {   // speculative prefetch 2 chunks ahead -> global_prefetch_b8
                __builtin_prefetch(A + (size_t)(m0 + rowA) * K + ((c + 2) << 5) + c4A, 0, 1);
                __builtin_prefetch(W + (size_t)(((c + 2) << 5) + rowW) * N + n0 + c4W, 0, 1);
            }
        }
        #pragma unroll
        for (int kk = 0; kk < 32; kk += 4) {
            v2f a = *(const v2f*)&As[cur][ar][kk + 2 * hf];
            const int pr = (kk >> 1) + hf;
            #pragma unroll
            for (int t = 0; t < 4; ++t) {
                v2f b = *(const v2f*)&WsP[cur][pr][2 * (t * 16 + l16)];
                acc[t] = WMMA4(a, b, acc[t]);
            }
        }
        if (more) STORE_TILE(cur ^ 1);
        __syncthreads();
    }

    const bool relu  = (flags & GF_RELU) != 0;
    const bool accum = (flags & GF_ACC) != 0;
    #pragma unroll
    for (int t = 0; t < 4; ++t) {
        const int n = n0 + t * 16 + l16;
        const float bs = bias ? bias[n] : 0.f;
        #pragma unroll
        for (int r = 0; r < 8; ++r) {
            int m = m0 + wave * 16 + r + hf * 8;
            size_t off = (size_t)m * N + n;
            float v0 = acc[t][r] + bs;
            if (accum) v0 += C[off];
            if (relu)  v0 = fmaxf(v0, 0.f);
            C[off] = v0;
        }
    }
#undef LOAD_REGS
#undef STORE_TILE
}

// ---------------------------------------------------------------------------
// Fused rel-pos attention (flash-style online softmax).
// Grid: (CS/64, B*H). Block: 256 threads / 8 waves (4 row-groups x 2 col
// halves); each block handles 64 query rows of one (b,h).
// rel_shift closed form: pos index t = j - i + CS-1; overflow (t > FS-1) is
// exactly the masked region j > i+PS.
// K^T and V tiles stored k-pair interleaved so B-frags are single b64 loads.
// ---------------------------------------------------------------------------
__launch_bounds__(256)
__global__ void attn_kernel(const float* __restrict__ q,
                            const float* __restrict__ kv,
                            const float* __restrict__ rp,
                            const float* __restrict__ up,
                            const float* __restrict__ vp,
                            float* __restrict__ out)
{
    __shared__ float Qu[64][68];
    __shared__ float Qv[64][68];
    __shared__ float Ktp[32][PSTRIDE];   // K^T pairs: [d>>1][2j + (d&1)]
    __shared__ float Vsp[32][PSTRIDE];   // V   pairs: [j>>1][2d + (j&1)]
    __shared__ float Sm[64][68];         // score / probability tile
    __shared__ float Rl[128][68];        // R band rows t = c0 .. c0+127
    __shared__ float m_run[64], l_run[64], alf[64];
    __shared__ float red[64][4];

    const int tid  = threadIdx.x;
    const int lane = tid & 31;
    const int wave = tid >> 5;
    const int wm   = wave & 3;
    const int wn   = wave >> 2;
    const int hf   = lane >> 4;
    const int l16  = lane & 15;
    const int it0  = blockIdx.x;
    const int i0   = it0 * 64;
    const int bh   = blockIdx.y;
    const int bb   = bh >> 4;       // / NH
    const int h    = bh & 15;

    // stage Q+u and Q+v
    {
        int row = tid >> 2;
        int d0  = (tid & 3) << 4;
        const float* qrow = q + ((size_t)(i0 + row) * BQ + bb) * DM + h * HDD;
        #pragma unroll
        for (int t = 0; t < 4; ++t) {
            int d = d0 + t * 4;
            float4 qq = *(const float4*)(qrow + d);
            float4 uu = *(const float4*)(up + h * HDD + d);
            float4 vv = *(const float4*)(vp + h * HDD + d);
            float4 a; a.x = qq.x + uu.x; a.y = qq.y + uu.y; a.z = qq.z + uu.z; a.w = qq.w + uu.w;
            float4 c; c.x = qq.x + vv.x; c.y = qq.y + vv.y; c.z = qq.z + vv.z; c.w = qq.w + vv.w;
            *(float4*)&Qu[row][d] = a;
            *(float4*)&Qv[row][d] = c;
        }
    }
    if (tid < 64) { m_run[tid] = -3.0e38f; l_run[tid] = 0.f; }
    v8f o0 = {0.f,0.f,0.f,0.f,0.f,0.f,0.f,0.f};
    v8f o1 = {0.f,0.f,0.f,0.f,0.f,0.f,0.f,0.f};
    __syncthreads();

    const int jt_end = it0 + 8;               // last chunk containing unmasked j
    for (int jt = 0; jt <= jt_end; ++jt) {
        const int j0 = jt * 64;
        const int c0 = j0 - i0 + 448;         // base row of R band

        // stage K (transposed, paired), V (paired), R band
        #pragma unroll
        for (int it = 0; it < 4; ++it) {
            int idx = tid + it * 256;
            int j   = idx >> 4;
            int d4  = (idx & 15) << 2;
            const float* kp = kv + ((size_t)(j0 + j) * BQ + bb) * (2 * DM) + h * HDD + d4;
            float4 kk = *(const float4*)kp;
            v2f t0; t0.x = kk.x; t0.y = kk.y;
            v2f t1; t1.x = kk.z; t1.y = kk.w;
            *(v2f*)&Ktp[(d4 >> 1) + 0][2 * j] = t0;
            *(v2f*)&Ktp[(d4 >> 1) + 1][2 * j] = t1;
            float4 vv = *(const float4*)(kp + DM);
            int jp = j >> 1, jb = j & 1;
            Vsp[jp][2 * (d4 + 0) + jb] = vv.x;
            Vsp[jp][2 * (d4 + 1) + jb] = vv.y;
            Vsp[jp][2 * (d4 + 2) + jb] = vv.z;
            Vsp[jp][2 * (d4 + 3) + jb] = vv.w;
        }
        #pragma unroll
        for (int it = 0; it < 8; ++it) {
            int idx = tid + it * 256;
            int s   = idx >> 4;
            int d4  = (idx & 15) << 2;
            int t   = c0 + s;
            float4 rr = make_float4(0.f, 0.f, 0.f, 0.f);
            if (t >= 0 && t < FSZ)
                rr = *(const float4*)(rp + (size_t)t * DM + h * HDD + d4);
            *(float4*)&Rl[s][d4] = rr;
        }
        __syncthreads();

        // content scores via WMMA: S = (Q+u) @ K^T
        v8f s0 = {0.f,0.f,0.f,0.f,0.f,0.f,0.f,0.f};
        v8f s1 = {0.f,0.f,0.f,0.f,0.f,0.f,0.f,0.f};
        {
            const int ar  = wm * 16 + l16;
            const int col = wn * 32 + l16;
            #pragma unroll
            for (int kk = 0; kk < 64; kk += 4) {
                v2f a = *(const v2f*)&Qu[ar][kk + 2 * hf];
                const int pr = (kk >> 1) + hf;
                v2f b0 = *(const v2f*)&Ktp[pr][2 * col];
                v2f b1 = *(const v2f*)&Ktp[pr][2 * (col + 16)];
                s0 = WMMA4(a, b0, s0);
                s1 = WMMA4(a, b1, s1);
            }
        }
        // add rel-pos term + scale + mask, write tile to LDS
        #pragma unroll
        for (int r = 0; r < 8; ++r) {
            int ml  = wm * 16 + r + hf * 8;
            int w0  = wn * 32 + l16;
            int w1  = w0 + 16;
            int sd0 = w0 - ml + 63;           // in [0,126]
            int sd1 = w1 - ml + 63;
            float p0 = 0.f, p1 = 0.f;
            #pragma unroll
            for (int d4 = 0; d4 < 64; d4 += 4) {
                float4 qv = *(const float4*)&Qv[ml][d4];
                float4 r0 = *(const float4*)&Rl[sd0][d4];
                float4 r1 = *(const float4*)&Rl[sd1][d4];
                p0 += qv.x * r0.x + qv.y * r0.y + qv.z * r0.z + qv.w * r0.w;
                p1 += qv.x * r1.x + qv.y * r1.y + qv.z * r1.z + qv.w * r1.w;
            }
            int i  = i0 + ml;
            int jA = j0 + w0, jB = j0 + w1;
            float t0 = (s0[r] + p0) * 0.125f;
            float t1 = (s1[r] + p1) * 0.125f;
            if (jA > i + PSZ) t0 = -1.0e30f;
            if (jB > i + PSZ) t1 = -1.0e30f;
            Sm[ml][w0] = t0;
            Sm[ml][w1] = t1;
        }
        __syncthreads();

        // row max partials
        {
            int row = tid >> 2, qq = tid & 3;
            float mx = -3.0e38f;
            #pragma unroll
            for (int w = 0; w < 16; ++w) mx = fmaxf(mx, Sm[row][qq * 16 + w]);
            red[row][qq] = mx;
        }
        __syncthreads();
        if (tid < 64) {
            float mx = fmaxf(fmaxf(red[tid][0], red[tid][1]),
                             fmaxf(red[tid][2], red[tid][3]));
            float mn = fmaxf(m_run[tid], mx);
            alf[tid] = __expf(m_run[tid] - mn);
            m_run[tid] = mn;
        }
        __syncthreads();
        // exponentiate + rowsum partials
        {
            int row = tid >> 2, qq = tid & 3;
            float mr = m_run[row];
            float sum = 0.f;
            #pragma unroll
            for (int w = 0; w < 16; ++w) {
                float p = __expf(Sm[row][qq * 16 + w] - mr);
                Sm[row][qq * 16 + w] = p;
                sum += p;
            }
            red[row][qq] = sum;
        }
        __syncthreads();
        if (tid < 64)
            l_run[tid] = alf[tid] * l_run[tid] +
                         red[tid][0] + red[tid][1] + red[tid][2] + red[tid][3];

        // rescale O, then O += P @ V via WMMA
        #pragma unroll
        for (int r = 0; r < 8; ++r) {
            float al = alf[wm * 16 + r + hf * 8];
            o0[r] *= al;
            o1[r] *= al;
        }
        {
            const int ar  = wm * 16 + l16;
            const int col = wn * 32 + l16;
            #pragma unroll
            for (int kk = 0; kk < 64; kk += 4) {
                v2f a = *(const v2f*)&Sm[ar][kk + 2 * hf];
                const int pr = (kk >> 1) + hf;
                v2f b0 = *(const v2f*)&Vsp[pr][2 * col];
                v2f b1 = *(const v2f*)&Vsp[pr][2 * (col + 16)];
                o0 = WMMA4(a, b0, o0);
                o1 = WMMA4(a, b1, o1);
            }
        }
        __syncthreads();
    }

    // normalize and write attn_vec (cs, b, H*HD)
    #pragma unroll
    for (int r = 0; r < 8; ++r) {
        int ml  = wm * 16 + r + hf * 8;
        float inv = 1.0f / l_run[ml];
        size_t base = ((size_t)(i0 + ml) * BQ + bb) * DM + h * HDD + wn * 32 + l16;
        out[base]      = o0[r] * inv;
        out[base + 16] = o1[r] * inv;
    }
}

// ---------------------------------------------------------------------------
// LayerNorm over last dim (1024). One block per row, float4 lanes.
// ---------------------------------------------------------------------------
__launch_bounds__(256)
__global__ void ln_kernel(const float* __restrict__ src,
                          const float* __restrict__ g,
                          const float* __restrict__ beta,
                          float* __restrict__ dst)
{
    __shared__ float red[256];
    const int row = blockIdx.x;
    const int c   = threadIdx.x << 2;
    float4 v = *(const float4*)(src + (size_t)row * DM + c);
    red[threadIdx.x] = v.x + v.y + v.z + v.w;
    __syncthreads();
    for (int off = 128; off > 0; off >>= 1) {
        if (threadIdx.x < off) red[threadIdx.x] += red[threadIdx.x + off];
        __syncthreads();
    }
    float mu = red[0] * (1.0f / DM);
    __syncthreads();
    float dx = v.x - mu, dy = v.y - mu, dz = v.z - mu, dw = v.w - mu;
    red[threadIdx.x] = dx * dx + dy * dy + dz * dz + dw * dw;
    __syncthreads();
    for (int off = 128; off > 0; off >>= 1) {
        if (threadIdx.x < off) red[threadIdx.x] += red[threadIdx.x + off];
        __syncthreads();
    }
    float rstd = rsqrtf(red[0] * (1.0f / DM) + 1e-5f);
    float4 gg = *(const float4*)(g + c);
    float4 bbv = *(const float4*)(beta + c);
    float4 o;
    o.x = dx * rstd * gg.x + bbv.x;
    o.y = dy * rstd * gg.y + bbv.y;
    o.z = dz * rstd * gg.z + bbv.z;
    o.w = dw * rstd * gg.w + bbv.w;
    *(float4*)(dst + (size_t)row * DM + c) = o;
}

// ---------------------------------------------------------------------------
// GRU-gate elementwise kernels (float4)
// ---------------------------------------------------------------------------
__device__ __forceinline__ float sigm(float x) { return 1.0f / (1.0f + __expf(-x)); }

__global__ void gate_rx_kernel(const float4* __restrict__ rpre,
                               const float4* __restrict__ x,
                               float4* __restrict__ out, int n4)
{
    int i = blockIdx.x * 256 + threadIdx.x;
    if (i < n4) {
        float4 r = rpre[i], xx = x[i], o;
        o.x = sigm(r.x) * xx.x; o.y = sigm(r.y) * xx.y;
        o.z = sigm(r.z) * xx.z; o.w = sigm(r.w) * xx.w;
        out[i] = o;
    }
}

__global__ void gate_out_kernel(const float4* __restrict__ zpre,
                                const float4* __restrict__ hpre,
                                const float4* __restrict__ x,
                                float4* __restrict__ out, int n4)
{
    int i = blockIdx.x * 256 + threadIdx.x;
    if (i < n4) {
        float4 zp = zpre[i], hp = hpre[i], xx = x[i], o;
        float z;
        z = sigm(zp.x - 0.2f); o.x = (1.f - z) * xx.x + z * tanhf(hp.x);
        z = sigm(zp.y - 0.2f); o.y = (1.f - z) * xx.y + z * tanhf(hp.y);
        z = sigm(zp.z - 0.2f); o.z = (1.f - z) * xx.z + z * tanhf(hp.z);
        z = sigm(zp.w - 0.2f); o.w = (1.f - z) * xx.w + z * tanhf(hp.w);
        out[i] = o;
    }
}

// ---------------------------------------------------------------------------
// Host orchestration
// ---------------------------------------------------------------------------
extern "C" void kernel_launch(void* const* d_in, const int* in_sizes, int n_in,
                              void* d_out, int out_size, void* d_ws, size_t ws_size,
                              hipStream_t stream)
{
    // Flattening assumption: recursive dict-insertion order of setup_inputs().
    const float* x_in   = (const float*)d_in[0];   // inputs (512,8,1024)
    const float* pose   = (const float*)d_in[1];   // pos_embedding (1024,1024)
    const float* mem    = (const float*)d_in[2];   // memory (512,8,1024)
    const float* u_p    = (const float*)d_in[3];
    const float* v_p    = (const float*)d_in[4];
    const float* Wkv    = (const float*)d_in[5];
    const float* bkv    = (const float*)d_in[6];
    const float* Wq     = (const float*)d_in[7];
    const float* bq     = (const float*)d_in[8];
    const float* Wo     = (const float*)d_in[9];
    const float* bo     = (const float*)d_in[10];
    const float* Wpos   = (const float*)d_in[11];
    const float* bpos   = (const float*)d_in[12];
    const float* ln_g   = (const float*)d_in[13];
    const float* ln_b   = (const float*)d_in[14];
    const float* g1Wr   = (const float*)d_in[15];
    const float* g1br   = (const float*)d_in[16];
    const float* g1Ur   = (const float*)d_in[17];
    const float* g1bur  = (const float*)d_in[18];
    const float* g1Wz   = (const float*)d_in[19];
    const float* g1bz   = (const float*)d_in[20];
    const float* g1Uz   = (const float*)d_in[21];
    const float* g1buz  = (const float*)d_in[22];
    const float* g1Wg   = (const float*)d_in[23];
    const float* g1bg   = (const float*)d_in[24];
    const float* g1Ug   = (const float*)d_in[25];
    const float* g1bug  = (const float*)d_in[26];
    const float* g2Wr   = (const float*)d_in[27];
    const float* g2br   = (const float*)d_in[28];
    const float* g2Ur   = (const float*)d_in[29];
    const float* g2bur  = (const float*)d_in[30];
    const float* g2Wz   = (const float*)d_in[31];
    const float* g2bz   = (const float*)d_in[32];
    const float* g2Uz   = (const float*)d_in[33];
    const float* g2buz  = (const float*)d_in[34];
    const float* g2Wg   = (const float*)d_in[35];
    const float* g2bg   = (const float*)d_in[36];
    const float* g2Ug   = (const float*)d_in[37];
    const float* g2bug  = (const float*)d_in[38];
    const float* W1     = (const float*)d_in[39];
    const float* b1     = (const float*)d_in[40];
    const float* W2     = (const float*)d_in[41];
    const float* b2     = (const float*)d_in[42];
    // d_in[43] = mask (recomputed analytically in the attention kernel)

    float* ws = (float*)d_ws;
    const size_t F = 1u << 20;                 // 1M floats
    float* X1 = ws;                            // 8F : LN(full_input)
    float* KV = ws + 8 * F;                    // 16F : kv  (later m1)
    float* Q  = ws + 24 * F;                   // 4F  : query (later Zg2)
    float* Rp = ws + 28 * F;                   // 1F  : pos projection
    float* AV = ws + 29 * F;                   // 4F  : attn_vec (later x2)
    float* A1 = ws + 33 * F;                   // 4F  : a1 (later o1)
    // reuse slots
    float* Rg  = X1;         float* Zg  = X1 + 4 * F;   // gate1 temps
    float* Hg  = Q;
    float* O1  = A1;
    float* X2  = AV;
    float* M1  = KV;
    float* M2  = X1;                                     // m2
    float* Rg2 = X1 + 4 * F; float* Zg2 = Q;
    float* Hg2 = KV;
    float* OUT = (float*)d_out;

    const int NM  = CSZ * BQ * DM;             // 4,194,304 elementwise size
    const int NM4 = NM / 4;

    auto gemm = [&](const float* A, const float* W, const float* bias, float* C,
                    int M, int N, int K, int flags) {
        dim3 grid(N / 64, M / 128);
        gemm_wmma_kernel<<<grid, 256, 0, stream>>>(A, W, bias, C, M, N, K, flags);
    };
    auto grx = [&](const float* r, const float* x, float* o) {
        gate_rx_kernel<<<NM4 / 256, 256, 0, stream>>>((const float4*)r, (const float4*)x,
                                                      (float4*)o, NM4);
    };
    auto gout = [&](const float* z, const float* h, const float* x, float* o) {
        gate_out_kernel<<<NM4 / 256, 256, 0, stream>>>((const float4*)z, (const float4*)h,
                                                       (const float4*)x, (float4*)o, NM4);
    };

    // 1. x1 = LN(concat(memory, inputs))
    ln_kernel<<<PSZ * BQ, 256, 0, stream>>>(mem,  ln_g, ln_b, X1);
    ln_kernel<<<CSZ * BQ, 256, 0, stream>>>(x_in, ln_g, ln_b, X1 + (size_t)PSZ * BQ * DM);
    // 2. projections
    gemm(X1,   Wkv,  bkv,  KV, FSZ * BQ, 2 * DM, DM, 0);
    gemm(x_in, Wq,   bq,   Q,  CSZ * BQ, DM,     DM, 0);
    gemm(pose, Wpos, bpos, Rp, FSZ,      DM,     DM, 0);
    // 3. fused rel-pos attention
    attn_kernel<<<dim3(CSZ / 64, BQ * NH), 256, 0, stream>>>(Q, KV, Rp, u_p, v_p, AV);
    // 4. a1 = relu(attn_vec @ Wo + bo)
    gemm(AV, Wo, bo, A1, CSZ * BQ, DM, DM, GF_RELU);
    // 5. gate1 (x = inputs, y = a1)
    gemm(A1,   g1Wr, g1br,  Rg, CSZ * BQ, DM, DM, 0);
    gemm(x_in, g1Ur, g1bur, Rg, CSZ * BQ, DM, DM, GF_ACC);
    gemm(A1,   g1Wz, g1bz,  Zg, CSZ * BQ, DM, DM, 0);
    gemm(x_in, g1Uz, g1buz, Zg, CSZ * BQ, DM, DM, GF_ACC);
    grx(Rg, x_in, Rg);                                   // rx
    gemm(A1, g1Wg, g1bg,  Hg, CSZ * BQ, DM, DM, 0);
    gemm(Rg, g1Ug, g1bug, Hg, CSZ * BQ, DM, DM, GF_ACC);
    gout(Zg, Hg, x_in, O1);                              // o1
    // 6. MLP
    ln_kernel<<<CSZ * BQ, 256, 0, stream>>>(O1, ln_g, ln_b, X2);
    gemm(X2, W1, b1, M1, CSZ * BQ, HID, DM,  GF_RELU);
    gemm(M1, W2, b2, M2, CSZ * BQ, DM,  HID, GF_RELU);
    // 7. gate2 (x = o1, y = m2)
    gemm(M2, g2Wr, g2br,  Rg2, CSZ * BQ, DM, DM, 0);
    gemm(O1, g2Ur, g2bur, Rg2, CSZ * BQ, DM, DM, GF_ACC);
    gemm(M2, g2Wz, g2bz,  Zg2, CSZ * BQ, DM, DM, 0);
    gemm(O1, g2Uz, g2buz, Zg2, CSZ * BQ, DM, DM, GF_ACC);
    grx(Rg2, O1, Rg2);
    gemm(M2,  g2Wg, g2bg,  Hg2, CSZ * BQ, DM, DM, 0);
    gemm(Rg2, g2Ug, g2bug, Hg2, CSZ * BQ, DM, DM, GF_ACC);
    gout(Zg2, Hg2, O1, OUT);                             // o2
}